// LogitsProcessor_77197742178950
// MI455X (gfx1250) — compile-verified
//
#include <hip/hip_runtime.h>

typedef __attribute__((ext_vector_type(2))) float v2f_t;
typedef __attribute__((ext_vector_type(4))) float f4_t;
typedef __attribute__((ext_vector_type(8))) float v8f_t;

#define BM 128
#define BN 128
#define BK 32
#define LDK 36                    // padded LDS k-stride (multiple of 4, even)
#define KMAX 8
#define NSPLIT 10
#define PSTRIDE (4 + 2 * KMAX)    // floats per (row,split) partial record
#define NEG_INF (-3.402823466e38f)

// ---------- wave32 16-lane-group reductions ----------
__device__ __forceinline__ float gmax16(float x) {
#pragma unroll
  for (int m = 1; m < 16; m <<= 1) x = fmaxf(x, __shfl_xor(x, m, 16));
  return x;
}
__device__ __forceinline__ float gsum16(float x) {
#pragma unroll
  for (int m = 1; m < 16; m <<= 1) x += __shfl_xor(x, m, 16);
  return x;
}

// ---------- kernel 0: last_index[b] = cumsum(seq_lens)-1 ----------
__global__ void k_prep(const int* __restrict__ seqlens, int Bn, int* __restrict__ lastidx) {
  if (threadIdx.x == 0 && blockIdx.x == 0) {
    int run = 0;
    for (int b = 0; b < Bn; ++b) { run += seqlens[b]; lastidx[b] = run - 1; }
  }
}

// ---------- kernel 1: fused GEMM (fp32 WMMA) + online softmax stats + top-k ----------
__global__ __launch_bounds__(256, 1)
void k_gemm_stats(const float* __restrict__ hidden,
                  const float* __restrict__ weight,
                  const int* __restrict__ input_ids,
                  const int* __restrict__ lastidx,
                  const int* __restrict__ topk_ptr,
                  int T, int D, int V, int Bn,
                  int chunks_per_split, int nchunks,
                  float* __restrict__ out,          // last_logits live at offset 0
                  float* __restrict__ part) {
  __shared__ __align__(16) float As[BM * LDK];
  __shared__ __align__(16) float Bs[BN * LDK];
  __shared__ float topv_s[BM][KMAX];
  __shared__ float topc_s[BM][KMAX];

  const int tid = threadIdx.x;
  const int lane = tid & 31;
  const int wid = tid >> 5;       // 8 waves, wave w owns rows [w*16, w*16+16)
  const int lm = lane & 15;       // N-lane within 16x16 tile
  const int hi = lane >> 4;       // row-half selector of C fragment
  const int row0 = blockIdx.x * BM;
  const int split = blockIdx.y;
  const int k = min(*topk_ptr, KMAX);

  if (tid < BM) {
#pragma unroll
    for (int j = 0; j < KMAX; ++j) { topv_s[tid][j] = NEG_INF; topc_s[tid][j] = -1.0f; }
  }

  // per-row constants for this lane's 8 rows (rows: row0 + wid*16 + 8*hi + v)
  int tgt[8]; int lastb[8];
#pragma unroll
  for (int v = 0; v < 8; ++v) {
    int row = row0 + wid * 16 + 8 * hi + v;
    tgt[v] = (row + 1 < T) ? input_ids[row + 1] : 0;
    int lb = -1;
    for (int b = 0; b < Bn; ++b) if (lastidx[b] == row) lb = b;
    lastb[v] = lb;
  }

  float mrow[8], srow[8], cap[8];
#pragma unroll
  for (int v = 0; v < 8; ++v) { mrow[v] = NEG_INF; srow[v] = 0.f; cap[v] = NEG_INF; }

  const int c_begin = split * chunks_per_split;
  const int c_end = min(nchunks, c_begin + chunks_per_split);

  // staging registers (double-buffer vs LDS: loads for step k+1 overlap WMMA of step k)
  const int srow_idx = tid >> 1;          // 0..127 (both A row and B vocab row)
  const int skh = (tid & 1) * 16;         // k-half within 32-wide tile
  f4_t ra[4], rb[4];

  for (int c = c_begin; c < c_end; ++c) {
    const int n0 = c * BN;
    v8f_t acc[8];
    const v8f_t vzero{};
#pragma unroll
    for (int s = 0; s < 8; ++s) acc[s] = vzero;

    // prologue: fetch first K-tile of this chunk into registers
    {
      const float* srcA = hidden + (size_t)(row0 + srow_idx) * D + skh;
#pragma unroll
      for (int i = 0; i < 4; ++i) ra[i] = *(const f4_t*)(srcA + 4 * i);
      const int gn = n0 + srow_idx;
      if (gn < V) {
        const float* srcB = weight + (size_t)gn * D + skh;
#pragma unroll
        for (int i = 0; i < 4; ++i) rb[i] = *(const f4_t*)(srcB + 4 * i);
      } else {
        const f4_t fz{};
#pragma unroll
        for (int i = 0; i < 4; ++i) rb[i] = fz;
      }
      // warm next vocab chunk's weight rows in L2 while we grind this one
      const int pn = n0 + BN + srow_idx;
      if (pn < V) __builtin_prefetch(weight + (size_t)pn * D + skh, 0, 0);
    }

    for (int k0 = 0; k0 < D; k0 += BK) {
      __syncthreads();  // previous tile's LDS reads complete
      {                 // commit staged registers to LDS
        float* dA = &As[srow_idx * LDK + skh];
        float* dB = &Bs[srow_idx * LDK + skh];
#pragma unroll
        for (int i = 0; i < 4; ++i) *(f4_t*)(dA + 4 * i) = ra[i];
#pragma unroll
        for (int i = 0; i < 4; ++i) *(f4_t*)(dB + 4 * i) = rb[i];
      }
      __syncthreads();

      // issue next tile's global loads BEFORE compute so WMMA hides their latency
      if (k0 + BK < D) {
        const int kn = k0 + BK;
        const float* srcA = hidden + (size_t)(row0 + srow_idx) * D + kn + skh;
#pragma unroll
        for (int i = 0; i < 4; ++i) ra[i] = *(const f4_t*)(srcA + 4 * i);
        const int gn = n0 + srow_idx;
        if (gn < V) {
          const float* srcB = weight + (size_t)gn * D + kn + skh;
#pragma unroll
          for (int i = 0; i < 4; ++i) rb[i] = *(const f4_t*)(srcB + 4 * i);
        } else {
          const f4_t fz{};
#pragma unroll
          for (int i = 0; i < 4; ++i) rb[i] = fz;
        }
      }

      // WMMA micro-kernel: A frag lane(m=lm, k=kk+2*hi), B frag lane(n=lm, k=kk+2*hi)
      const int abase = (wid * 16 + lm) * LDK + 2 * hi;
#pragma unroll
      for (int kk = 0; kk < BK; kk += 4) {
        v2f_t a = *(const v2f_t*)&As[abase + kk];
#pragma unroll
        for (int s = 0; s < 8; ++s) {
          v2f_t b = *(const v2f_t*)&Bs[(s * 16 + lm) * LDK + 2 * hi + kk];
          acc[s] = __builtin_amdgcn_wmma_f32_16x16x4_f32(
              false, a, false, b, (short)0, acc[s], false, false);
        }
      }
    }

    // ---- chunk epilogue: online softmax + running top-k + captures ----
#pragma unroll
    for (int v = 0; v < 8; ++v) {
      const int rl = wid * 16 + 8 * hi + v;
      float x[8];
      float lmax = NEG_INF;
      float cp = cap[v];
#pragma unroll
      for (int s = 0; s < 8; ++s) {
        int col = n0 + s * 16 + lm;
        float xv = acc[s][v];
        if (col >= V) xv = NEG_INF;
        x[s] = xv;
        if (col < V && col == tgt[v]) cp = xv;
        lmax = fmaxf(lmax, xv);
      }
      cap[v] = cp;

      if (lastb[v] >= 0) {  // stream last-row logits straight to output
#pragma unroll
        for (int s = 0; s < 8; ++s) {
          int col = n0 + s * 16 + lm;
          if (col < V) out[(size_t)lastb[v] * V + col] = x[s];
        }
      }

      float gm = gmax16(lmax);
      if (gm > -1e37f) {
        float mnew = fmaxf(mrow[v], gm);
        float e = 0.f;
#pragma unroll
        for (int s = 0; s < 8; ++s) e += __expf(x[s] - mnew);
        e = gsum16(e);
        srow[v] = srow[v] * __expf(mrow[v] - mnew) + e;
        mrow[v] = mnew;
      }

      float tmin = topv_s[rl][k - 1];
      if (gm > tmin) {  // chunk can contribute to top-k
        float tv[8];
#pragma unroll
        for (int s = 0; s < 8; ++s) tv[s] = x[s];
        for (int j = 0; j < k; ++j) {
          float bv = NEG_INF; int bc = 0x7fffffff;
#pragma unroll
          for (int s = 0; s < 8; ++s) {
            int col = n0 + s * 16 + lm;
            if (tv[s] > bv || (tv[s] == bv && col < bc)) { bv = tv[s]; bc = col; }
          }
#pragma unroll
          for (int msk = 1; msk < 16; msk <<= 1) {
            float ov = __shfl_xor(bv, msk, 16);
            int oc = __shfl_xor(bc, msk, 16);
            if (ov > bv || (ov == bv && oc < bc)) { bv = ov; bc = oc; }
          }
#pragma unroll
          for (int s = 0; s < 8; ++s) {
            int col = n0 + s * 16 + lm;
            if (col == bc) tv[s] = NEG_INF;
          }
          if (lm == 0) {  // group leader inserts into sorted LDS list
            if (bv > topv_s[rl][k - 1]) {
              int p = k - 1;
              while (p > 0 && topv_s[rl][p - 1] < bv) {
                topv_s[rl][p] = topv_s[rl][p - 1];
                topc_s[rl][p] = topc_s[rl][p - 1];
                --p;
              }
              topv_s[rl][p] = bv;
              topc_s[rl][p] = (float)bc;
            }
          }
        }
      }
    }
  }

  // ---- write per-(row,split) partials ----
  const int nsplit = gridDim.y;
  if (lm == 0) {
#pragma unroll
    for (int v = 0; v < 8; ++v) {
      int row = row0 + wid * 16 + 8 * hi + v;
      float* p = part + ((size_t)row * nsplit + split) * PSTRIDE;
      p[0] = mrow[v]; p[1] = srow[v]; p[2] = cap[v];
    }
  }
  __syncthreads();
  if (tid < BM) {
    int row = row0 + tid;
    float* p = part + ((size_t)row * nsplit + split) * PSTRIDE;
#pragma unroll
    for (int j = 0; j < KMAX; ++j) { p[4 + j] = topv_s[tid][j]; p[4 + KMAX + j] = topc_s[tid][j]; }
  }
}

// ---------- kernel 2: merge splits per row -> lse, prefill logprob, top-k ----------
__global__ void k_merge(const float* __restrict__ part,
                        const int* __restrict__ topk_ptr,
                        int T, int V, int Bn, int nsplit,
                        float* __restrict__ out,
                        float* __restrict__ lse_ws) {
  int r = blockIdx.x * blockDim.x + threadIdx.x;
  if (r >= T) return;
  int k = min(*topk_ptr, KMAX);
  size_t BV = (size_t)Bn * V;
  size_t off_prefill = 2 * BV + Bn;
  size_t off_tv = off_prefill + T;
  size_t off_ti = off_tv + (size_t)T * k;

  const float* pr = part + (size_t)r * nsplit * PSTRIDE;
  float m = NEG_INF, cap = NEG_INF;
  for (int i = 0; i < nsplit; ++i) m = fmaxf(m, pr[i * PSTRIDE]);
  float ssum = 0.f;
  for (int i = 0; i < nsplit; ++i) {
    ssum += pr[i * PSTRIDE + 1] * __expf(pr[i * PSTRIDE] - m);
    cap = fmaxf(cap, pr[i * PSTRIDE + 2]);
  }
  float lse = m + __logf(ssum);
  lse_ws[r] = lse;
  out[off_prefill + r] = cap - lse;

  float cv[NSPLIT * KMAX];
  float cc[NSPLIT * KMAX];
  int n = 0;
  for (int i = 0; i < nsplit; ++i)
    for (int j = 0; j < k; ++j) {
      cv[n] = pr[i * PSTRIDE + 4 + j];
      cc[n] = pr[i * PSTRIDE + 4 + KMAX + j];
      ++n;
    }
  for (int j = 0; j < k; ++j) {
    float bv = NEG_INF, bc = 3.0e9f; int bi = 0;
    for (int i = 0; i < n; ++i)
      if (cv[i] > bv || (cv[i] == bv && cc[i] < bc)) { bv = cv[i]; bc = cc[i]; bi = i; }
    out[off_tv + (size_t)r * k + j] = bv - lse;
    out[off_ti + (size_t)r * k + j] = bc;
    cv[bi] = NEG_INF;
  }
}

// ---------- kernel 3: last_logprobs = last_logits - lse[last_index[b]] ----------
__global__ void k_lastlp(const float* __restrict__ lse_ws, const int* __restrict__ lastidx,
                         int V, int Bn, float* __restrict__ out) {
  size_t idx = (size_t)blockIdx.x * blockDim.x + threadIdx.x;
  size_t BV = (size_t)Bn * V;
  if (idx >= BV) return;
  int b = (int)(idx / V);
  out[BV + idx] = out[idx] - lse_ws[lastidx[b]];
}

// ---------- kernel 4: decode top-k gather + normalized prompt logprobs ----------
__global__ void k_final(const int* __restrict__ lastidx,
                        const int* __restrict__ seqlens, const int* __restrict__ startloc,
                        const int* __restrict__ topk_ptr,
                        int T, int V, int Bn, float* __restrict__ out) {
  __shared__ float red[256];
  int tid = threadIdx.x;
  int k = min(*topk_ptr, KMAX);
  size_t BV = (size_t)Bn * V;
  size_t off_norm = 2 * BV;
  size_t off_prefill = off_norm + Bn;
  size_t off_tv = off_prefill + T;
  size_t off_ti = off_tv + (size_t)T * k;
  size_t off_dv = off_ti + (size_t)T * k;
  size_t off_di = off_dv + (size_t)Bn * k;

  if (tid < Bn * k) {
    int b = tid / k, j = tid % k;
    int lr = lastidx[b];
    out[off_dv + tid] = out[off_tv + (size_t)lr * k + j];
    out[off_di + tid] = out[off_ti + (size_t)lr * k + j];
  }
  for (int b = 0; b < Bn; ++b) {
    int start = startloc[b];
    int len = seqlens[b];
    int s0 = min(max(start, 0), T - 1);
    int e0 = min(max(start + len - 2, 0), T - 1);
    float a = 0.f;
    for (int t = s0 + tid; t <= e0; t += blockDim.x) a += out[off_prefill + t];
    red[tid] = a;
    __syncthreads();
    for (int st = 128; st > 0; st >>= 1) {
      if (tid < st) red[tid] += red[tid + st];
      __syncthreads();
    }
    if (tid == 0) out[off_norm + b] = red[0] / fmaxf((float)(len - 1), 1.0f);
    __syncthreads();
  }
}

extern "C" void kernel_launch(void* const* d_in, const int* in_sizes, int n_in,
                              void* d_out, int out_size, void* d_ws, size_t ws_size,
                              hipStream_t stream) {
  const int* input_ids = (const int*)d_in[0];
  const float* hidden = (const float*)d_in[1];
  const float* weight = (const float*)d_in[2];
  const int* seqlens = (const int*)d_in[3];
  const int* startloc = (const int*)d_in[4];
  const int* topk = (const int*)d_in[5];

  int T = in_sizes[0];
  int D = in_sizes[1] / T;
  int V = in_sizes[2] / D;
  int Bn = in_sizes[3];
  float* out = (float*)d_out;

  int* lastidx = (int*)d_ws;                       // B ints (16-float slot)
  float* part = (float*)d_ws + 16;                 // T*NSPLIT*PSTRIDE floats
  float* lse_ws = part + (size_t)T * NSPLIT * PSTRIDE;  // T floats

  k_prep<<<1, 1, 0, stream>>>(seqlens, Bn, lastidx);

  int nchunks = (V + BN - 1) / BN;
  int cps = (nchunks + NSPLIT - 1) / NSPLIT;
  dim3 g1(T / BM, NSPLIT);
  k_gemm_stats<<<g1, 256, 0, stream>>>(hidden, weight, input_ids, lastidx, topk,
                                       T, D, V, Bn, cps, nchunks, out, part);
  k_merge<<<(T + 255) / 256, 256, 0, stream>>>(part, topk, T, V, Bn, NSPLIT, out, lse_ws);
  size_t BV = (size_t)Bn * V;
  k_lastlp<<<(int)((BV + 255) / 256), 256, 0, stream>>>(lse_ws, lastidx, V, Bn, out);
  k_final<<<1, 256, 0, stream>>>(lastidx, seqlens, startloc, topk, T, V, Bn, out);
}